// DecodePredictions_7842610283045
// MI455X (gfx1250) — compile-verified
//
#include <hip/hip_runtime.h>
#include <math.h>
#include <stdint.h>

// ---------------------------------------------------------------------------
// RetinaNet DecodePredictions post-processing for MI455X (gfx1250, wave32).
// HBM-bound: two TDM-staged streaming passes over the 196MB logit tensor.
// Top-k threshold search uses v_wmma_f32_16x16x4_f32 prefix-scan.
// ---------------------------------------------------------------------------

#define B_SZ      8
#define A_TOTAL   76725
#define NCLS      80
#define NBINS     128
#define PRE_K     1000
#define M_PER     100
#define MAX_DET   100
#define SEG_ROWS  512
#define TILE_ROWS 32
#define NSEG      ((A_TOTAL + SEG_ROWS - 1) / SEG_ROWS)   // 150
#define TILES_PER_SEG (SEG_ROWS / TILE_ROWS)              // 16
#define SCORE_LOGIT_NEG_INF (-INFINITY)

typedef float v2f __attribute__((ext_vector_type(2)));
typedef float v8f __attribute__((ext_vector_type(8)));
typedef unsigned int v4u __attribute__((ext_vector_type(4)));
typedef int v8i_t __attribute__((ext_vector_type(8)));
typedef int v4i_t __attribute__((ext_vector_type(4)));

// ----------------------- TDM: async global tile -> LDS ----------------------
// Stages a [TILE_ROWS x 80] f32 tile (contiguous rows, stride 80) into LDS.
// rows_avail = remaining valid rows; TDM zero-fills reads past tensor_dim1.
__device__ __forceinline__ void tdm_load_tile(const float* gptr,
                                              unsigned lds_byte_off,
                                              unsigned rows_avail) {
  uint64_t ga = (uint64_t)(uintptr_t)gptr;
  v4u g0;
  g0[0] = 1u;                                            // count=1, no gather
  g0[1] = lds_byte_off;                                  // lds_addr
  g0[2] = (unsigned)ga;                                  // global_addr lo
  g0[3] = ((unsigned)(ga >> 32) & 0x01FFFFFFu) | (2u << 30); // addr hi | type=2
  v8i_t g1;
  g1[0] = (int)(2u << 16);                               // data_size = 4B
  g1[1] = (int)(80u << 16);                              // tensor_dim0 lo16
  g1[2] = (int)((rows_avail & 0xFFFFu) << 16);           // dim0 hi | dim1 lo
  g1[3] = (int)((rows_avail >> 16) | (80u << 16));       // dim1 hi | tile_dim0
  g1[4] = (int)TILE_ROWS;                                // tile_dim1, tile_dim2=0
  g1[5] = 80;                                            // dim0_stride lo32
  g1[6] = 0;
  g1[7] = 0;
  v4i_t gz4 = {0, 0, 0, 0};
  v8i_t gz8 = {0, 0, 0, 0, 0, 0, 0, 0};
  __builtin_amdgcn_tensor_load_to_lds(g0, g1, gz4, gz4, gz8, 0);
}

__device__ __forceinline__ unsigned lds_off_of(const void* p) {
  return (unsigned)(uintptr_t)p;   // low 32 bits = LDS byte offset (aperture)
}

// Order-preserving map: float bits -> unsigned (ascending), then descending bin
__device__ __forceinline__ unsigned score_bin_desc(float x) {
  unsigned u = __float_as_uint(x);
  unsigned ord = (u & 0x80000000u) ? ~u : (u | 0x80000000u);
  return 127u - (ord >> 25);       // bin 0 = highest logits
}

// ----------------------------- K1: box decode -------------------------------
__global__ __launch_bounds__(256) void decode_boxes_kernel(
    const float* __restrict__ boxp, const float* __restrict__ anch,
    float* __restrict__ boxes, int n) {
  int t = blockIdx.x * blockDim.x + threadIdx.x;
  if (t >= n) return;
  int a = t % A_TOTAL;
  float4 bp = ((const float4*)boxp)[t];
  float4 an = ((const float4*)anch)[a];
  float tx = bp.x * 0.1f, ty = bp.y * 0.1f;
  float tw = bp.z * 0.2f, th = bp.w * 0.2f;
  float cx = tx * an.z + an.x;
  float cy = ty * an.w + an.y;
  float w  = __expf(tw) * an.z;
  float h  = __expf(th) * an.w;
  const float inv = 1.0f / 640.0f;
  float4 o;
  o.x = (cx - 0.5f * w) * inv; o.y = (cy - 0.5f * h) * inv;
  o.z = (cx + 0.5f * w) * inv; o.w = (cy + 0.5f * h) * inv;
  ((float4*)boxes)[t] = o;
}

// ------------------- K2: TDM-staged per-class histograms --------------------
__global__ __launch_bounds__(256) void class_hist_kernel(
    const float* __restrict__ cls_logits, unsigned* __restrict__ ghist) {
  __shared__ __align__(16) float stage[2][TILE_ROWS * 80];
  __shared__ unsigned lhist[NCLS * NBINS];
  const int seg = blockIdx.x, b = blockIdx.y, t = threadIdx.x;
  for (int i = t; i < NCLS * NBINS; i += 256) lhist[i] = 0u;

  const int row0 = seg * SEG_ROWS;
  const float* base = cls_logits + (size_t)b * A_TOTAL * 80;

  if (t < 32) {  // wave 0 drives the TDM pipeline
    int r = row0 < (A_TOTAL - 1) ? row0 : (A_TOTAL - 1);
    tdm_load_tile(base + (size_t)r * 80, lds_off_of(&stage[0][0]),
                  (unsigned)(A_TOTAL - r));
  }
  for (int tt = 0; tt < TILES_PER_SEG; ++tt) {
    if (t < 32) {
      if (tt + 1 < TILES_PER_SEG) {
        int r = row0 + (tt + 1) * TILE_ROWS;
        int rc = r < (A_TOTAL - 1) ? r : (A_TOTAL - 1);
        tdm_load_tile(base + (size_t)rc * 80,
                      lds_off_of(&stage[(tt + 1) & 1][0]),
                      (unsigned)(A_TOTAL - rc));
        __builtin_amdgcn_s_wait_tensorcnt(1);   // current tile complete
      } else {
        __builtin_amdgcn_s_wait_tensorcnt(0);
      }
    }
    __syncthreads();
    const float* buf = stage[tt & 1];
    const int rbase = row0 + tt * TILE_ROWS;
    for (int e = t; e < TILE_ROWS * 80; e += 256) {
      int r = e / 80, c = e - r * 80;
      if (rbase + r < A_TOTAL) {
        unsigned bin = score_bin_desc(buf[e]);
        atomicAdd(&lhist[c * NBINS + bin], 1u);
      }
    }
    __syncthreads();   // tile consumed before its buffer is re-filled
  }
  for (int i = t; i < NCLS * NBINS; i += 256) {
    unsigned v = lhist[i];
    if (v) atomicAdd(&ghist[(size_t)b * NCLS * NBINS + i], v);
  }
}

// ---------- K3: WMMA prefix-scan over 128-bin hist -> top-K threshold -------
// 256-slot inclusive prefix P = H*U + Lstrict*T on the matrix pipe:
//   S1 = H x U          (row-inclusive prefix, 4x wmma K=4)
//   P  = S1 + Lstrict x T1  (cross-row carry, 4x wmma K=4), T1 row M = total_M
__global__ __launch_bounds__(32) void topk_threshold_kernel(
    const unsigned* __restrict__ ghist, unsigned* __restrict__ thrbin) {
  __shared__ float h[256];
  __shared__ float totals[16];
  __shared__ unsigned redmin[32];
  const int bc = blockIdx.x, lane = threadIdx.x;
  const unsigned* hbc = ghist + (size_t)bc * NBINS;
  for (int i = lane; i < 256; i += 32) h[i] = (i < NBINS) ? (float)hbc[i] : 0.0f;
  __syncthreads();

  const int half = lane >> 4;     // lanes 16-31 hold second K/M halves
  const int l16 = lane & 15;
  v8f acc = {};
  // S1 = H x U   (U[k][n] = k<=n)
#pragma unroll
  for (int q = 0; q < 4; ++q) {
    int k0 = 4 * q + 2 * half;
    v2f a, bfr;
    a.x = h[l16 * 16 + k0 + 0];
    a.y = h[l16 * 16 + k0 + 1];
    bfr.x = ((k0 + 0) <= l16) ? 1.0f : 0.0f;
    bfr.y = ((k0 + 1) <= l16) ? 1.0f : 0.0f;
    acc = __builtin_amdgcn_wmma_f32_16x16x4_f32(false, a, false, bfr,
                                                (short)0, acc, false, false);
  }
  // row totals live in column 15 -> lanes 15 / 31
  if (l16 == 15) {
#pragma unroll
    for (int r = 0; r < 8; ++r) totals[r + 8 * half] = acc[r];
  }
  __syncthreads();
  // P = S1 + Lstrict x T1   (Lstrict[M][k] = k<M ; T1[k][*] = totals[k])
#pragma unroll
  for (int q = 0; q < 4; ++q) {
    int k0 = 4 * q + 2 * half;
    v2f a, bfr;
    a.x = ((k0 + 0) < l16) ? 1.0f : 0.0f;
    a.y = ((k0 + 1) < l16) ? 1.0f : 0.0f;
    bfr.x = totals[k0 + 0];
    bfr.y = totals[k0 + 1];
    acc = __builtin_amdgcn_wmma_f32_16x16x4_f32(false, a, false, bfr,
                                                (short)0, acc, false, false);
  }
  // smallest descending-bin index whose inclusive count >= PRE_K
  unsigned best = NBINS - 1;
#pragma unroll
  for (int r = 0; r < 8; ++r) {
    int M = r + 8 * half;
    int flat = M * 16 + l16;
    if (flat < NBINS && acc[r] >= (float)PRE_K)
      best = min(best, (unsigned)flat);
  }
  redmin[lane] = best;
  __syncthreads();
  for (int s = 16; s > 0; s >>= 1) {
    if (lane < s) redmin[lane] = min(redmin[lane], redmin[lane + s]);
    __syncthreads();
  }
  if (lane == 0) thrbin[bc] = redmin[0];
}

// -------------- K4: TDM-staged compaction of top-K candidates ---------------
__global__ __launch_bounds__(256) void compact_kernel(
    const float* __restrict__ cls_logits, const unsigned* __restrict__ thrbin,
    unsigned* __restrict__ cnt, unsigned* __restrict__ cidx,
    float* __restrict__ clogit) {
  __shared__ __align__(16) float stage[2][TILE_ROWS * 80];
  __shared__ unsigned sthr[NCLS];
  const int seg = blockIdx.x, b = blockIdx.y, t = threadIdx.x;
  if (t < NCLS) sthr[t] = thrbin[b * NCLS + t];

  const int row0 = seg * SEG_ROWS;
  const float* base = cls_logits + (size_t)b * A_TOTAL * 80;
  if (t < 32) {
    int r = row0 < (A_TOTAL - 1) ? row0 : (A_TOTAL - 1);
    tdm_load_tile(base + (size_t)r * 80, lds_off_of(&stage[0][0]),
                  (unsigned)(A_TOTAL - r));
  }
  for (int tt = 0; tt < TILES_PER_SEG; ++tt) {
    if (t < 32) {
      if (tt + 1 < TILES_PER_SEG) {
        int r = row0 + (tt + 1) * TILE_ROWS;
        int rc = r < (A_TOTAL - 1) ? r : (A_TOTAL - 1);
        tdm_load_tile(base + (size_t)rc * 80,
                      lds_off_of(&stage[(tt + 1) & 1][0]),
                      (unsigned)(A_TOTAL - rc));
        __builtin_amdgcn_s_wait_tensorcnt(1);
      } else {
        __builtin_amdgcn_s_wait_tensorcnt(0);
      }
    }
    __syncthreads();
    const float* buf = stage[tt & 1];
    const int rbase = row0 + tt * TILE_ROWS;
    for (int e = t; e < TILE_ROWS * 80; e += 256) {
      int r = e / 80, c = e - r * 80;
      int anchor = rbase + r;
      if (anchor < A_TOTAL) {
        float x = buf[e];
        if (score_bin_desc(x) <= sthr[c]) {
          unsigned pos = atomicAdd(&cnt[b * NCLS + c], 1u);
          if (pos < PRE_K) {
            size_t o = (size_t)(b * NCLS + c) * PRE_K + pos;
            cidx[o] = (unsigned)anchor;
            clogit[o] = x;
          }
        }
      }
    }
    __syncthreads();
  }
}

// ------------------------- K5: per-(b,c) greedy NMS -------------------------
__global__ __launch_bounds__(128) void nms_kernel(
    const float* __restrict__ boxes, const unsigned* __restrict__ cnt,
    const unsigned* __restrict__ cidx, const float* __restrict__ clogit,
    float* __restrict__ nmsout) {
  __shared__ float bx1[PRE_K], by1[PRE_K], bx2[PRE_K], by2[PRE_K];
  __shared__ float sc[PRE_K], ar[PRE_K];
  __shared__ float rv[128];
  __shared__ int ri[128];
  __shared__ float selbox[4];
  __shared__ int seli, hasflag;

  const int bc = blockIdx.x, b = bc / NCLS, t = threadIdx.x;
  const int nc = (int)min(cnt[bc], (unsigned)PRE_K);
  for (int i = t; i < PRE_K; i += 128) {
    if (i < nc) {
      unsigned a = cidx[(size_t)bc * PRE_K + i];
      const float* bp = boxes + ((size_t)b * A_TOTAL + a) * 4;
      float x1 = bp[0], y1 = bp[1], x2 = bp[2], y2 = bp[3];
      bx1[i] = x1; by1[i] = y1; bx2[i] = x2; by2[i] = y2;
      ar[i] = (x2 - x1) * (y2 - y1);
      sc[i] = clogit[(size_t)bc * PRE_K + i];
    } else {
      sc[i] = SCORE_LOGIT_NEG_INF;
    }
  }
  __syncthreads();

  float* out = nmsout + (size_t)bc * M_PER * 6;
  for (int m = 0; m < M_PER; ++m) {
    float bv = SCORE_LOGIT_NEG_INF;
    int bi = 0;
    for (int i = t; i < nc; i += 128) {
      float v = sc[i];
      if (v > bv) { bv = v; bi = i; }
    }
    rv[t] = bv; ri[t] = bi;
    __syncthreads();
    for (int s = 64; s > 0; s >>= 1) {
      if (t < s && rv[t + s] > rv[t]) { rv[t] = rv[t + s]; ri[t] = ri[t + s]; }
      __syncthreads();
    }
    if (t == 0) {
      int i = ri[0];
      float v = rv[0];
      int has = (nc > 0) && (v > SCORE_LOGIT_NEG_INF);
      float x1 = has ? bx1[i] : 0.f, y1 = has ? by1[i] : 0.f;
      float x2 = has ? bx2[i] : 0.f, y2 = has ? by2[i] : 0.f;
      float s = has ? 1.0f / (1.0f + __expf(-v)) : 0.0f;
      out[m * 6 + 0] = x1; out[m * 6 + 1] = y1;
      out[m * 6 + 2] = x2; out[m * 6 + 3] = y2;
      out[m * 6 + 4] = s;
      out[m * 6 + 5] = (s > 0.05f) ? 1.0f : 0.0f;
      selbox[0] = x1; selbox[1] = y1; selbox[2] = x2; selbox[3] = y2;
      seli = i; hasflag = has;
      if (has) sc[i] = SCORE_LOGIT_NEG_INF;
    }
    __syncthreads();
    if (hasflag) {
      float X1 = selbox[0], Y1 = selbox[1], X2 = selbox[2], Y2 = selbox[3];
      float sa = (X2 - X1) * (Y2 - Y1);
      for (int i = t; i < nc; i += 128) {
        float ix1 = fmaxf(X1, bx1[i]), iy1 = fmaxf(Y1, by1[i]);
        float ix2 = fminf(X2, bx2[i]), iy2 = fminf(Y2, by2[i]);
        float iw = fmaxf(ix2 - ix1, 0.f), ih = fmaxf(iy2 - iy1, 0.f);
        float inter = iw * ih;
        float iou = inter / (sa + ar[i] - inter + 1e-8f);
        if (iou > 0.5f) sc[i] = SCORE_LOGIT_NEG_INF;
      }
    }
    __syncthreads();
  }
}

// -------------------- K6: per-batch final top-100 merge ---------------------
__global__ __launch_bounds__(256) void final_topk_kernel(
    const float* __restrict__ nmsout, float* __restrict__ out_boxes,
    float* __restrict__ out_scores, float* __restrict__ out_cls,
    int* __restrict__ out_valid) {
  __shared__ float s[NCLS * M_PER];   // 8000 floats
  __shared__ float rv[256];
  __shared__ int ri[256];
  __shared__ int validcnt;
  const int b = blockIdx.x, t = threadIdx.x;
  if (t == 0) validcnt = 0;
  const float* nb = nmsout + (size_t)b * NCLS * M_PER * 6;
  for (int i = t; i < NCLS * M_PER; i += 256)
    s[i] = (nb[i * 6 + 5] > 0.5f) ? nb[i * 6 + 4] : -1.0f;
  __syncthreads();

  for (int m = 0; m < MAX_DET; ++m) {
    float bv = -3.0f;
    int bi = 0;
    for (int i = t; i < NCLS * M_PER; i += 256) {
      float v = s[i];
      if (v > bv) { bv = v; bi = i; }
    }
    rv[t] = bv; ri[t] = bi;
    __syncthreads();
    for (int st = 128; st > 0; st >>= 1) {
      if (t < st && rv[t + st] > rv[t]) { rv[t] = rv[t + st]; ri[t] = ri[t + st]; }
      __syncthreads();
    }
    if (t == 0) {
      int i = ri[0];
      float v = rv[0];
      int valid = v > -0.5f;
      size_t ob = (size_t)(b * MAX_DET + m);
      out_boxes[ob * 4 + 0] = valid ? nb[i * 6 + 0] : 0.f;
      out_boxes[ob * 4 + 1] = valid ? nb[i * 6 + 1] : 0.f;
      out_boxes[ob * 4 + 2] = valid ? nb[i * 6 + 2] : 0.f;
      out_boxes[ob * 4 + 3] = valid ? nb[i * 6 + 3] : 0.f;
      out_scores[ob] = valid ? v : 0.f;
      out_cls[ob] = valid ? (float)(i / M_PER) : 0.f;
      if (valid) validcnt++;
      s[i] = -2.0f;
    }
    __syncthreads();
  }
  if (t == 0) out_valid[b] = validcnt;
}

// --------------------------------- launch -----------------------------------
extern "C" void kernel_launch(void* const* d_in, const int* in_sizes, int n_in,
                              void* d_out, int out_size, void* d_ws,
                              size_t ws_size, hipStream_t stream) {
  const float* boxp = (const float*)d_in[0];   // [8,76725,4]
  const float* clsp = (const float*)d_in[1];   // [8,76725,80]
  const float* anch = (const float*)d_in[2];   // [76725,4]
  float* out = (float*)d_out;                  // fb|fs|fc|valid flat

  char* ws = (char*)d_ws;
  size_t off = 0;
  auto take = [&](size_t bytes) -> void* {
    void* p = ws + off;
    off = (off + bytes + 255) & ~(size_t)255;
    return p;
  };
  float*    w_boxes  = (float*)take((size_t)B_SZ * A_TOTAL * 4 * sizeof(float));
  unsigned* w_hist   = (unsigned*)take((size_t)B_SZ * NCLS * NBINS * sizeof(unsigned));
  unsigned* w_thr    = (unsigned*)take((size_t)B_SZ * NCLS * sizeof(unsigned));
  unsigned* w_cnt    = (unsigned*)take((size_t)B_SZ * NCLS * sizeof(unsigned));
  unsigned* w_cidx   = (unsigned*)take((size_t)B_SZ * NCLS * PRE_K * sizeof(unsigned));
  float*    w_clogit = (float*)take((size_t)B_SZ * NCLS * PRE_K * sizeof(float));
  float*    w_nms    = (float*)take((size_t)B_SZ * NCLS * M_PER * 6 * sizeof(float));

  (void)hipMemsetAsync(w_hist, 0,
                       (size_t)B_SZ * NCLS * NBINS * sizeof(unsigned), stream);
  (void)hipMemsetAsync(w_cnt, 0, (size_t)B_SZ * NCLS * sizeof(unsigned),
                       stream);

  const int n_boxes = B_SZ * A_TOTAL;
  decode_boxes_kernel<<<(n_boxes + 255) / 256, 256, 0, stream>>>(
      boxp, anch, w_boxes, n_boxes);
  class_hist_kernel<<<dim3(NSEG, B_SZ), 256, 0, stream>>>(clsp, w_hist);
  topk_threshold_kernel<<<B_SZ * NCLS, 32, 0, stream>>>(w_hist, w_thr);
  compact_kernel<<<dim3(NSEG, B_SZ), 256, 0, stream>>>(clsp, w_thr, w_cnt,
                                                       w_cidx, w_clogit);
  nms_kernel<<<B_SZ * NCLS, 128, 0, stream>>>(w_boxes, w_cnt, w_cidx,
                                              w_clogit, w_nms);
  final_topk_kernel<<<B_SZ, 256, 0, stream>>>(
      w_nms, out, out + B_SZ * MAX_DET * 4,
      out + B_SZ * MAX_DET * 5, (int*)(out + B_SZ * MAX_DET * 6));
}